// ModelNew_3556232921911
// MI455X (gfx1250) — compile-verified
//
#include <hip/hip_runtime.h>

typedef __attribute__((ext_vector_type(2))) float v2f;
typedef __attribute__((ext_vector_type(8))) float v8f;

#define IN_C   3
#define OUT_C  16
#define D_IN   16
#define H_IN   32
#define W_IN   32
#define D_OUT  31
#define H_OUT  63
#define W_OUT  63

// LW: repacked weights [kd*3+kh][k = ic*3+kw, padded to 12][oc = 16]
#define LW_SZ (9 * 12 * 16)

__global__ __launch_bounds__(128) void convt_lse_hswish_kernel(
    const float* __restrict__ x,      // (32,3,16,32,32)
    const float* __restrict__ w,      // (3,16,3,3,3): ic*432 + oc*27 + kd*9 + kh*3 + kw
    const float* __restrict__ bias,   // scalar
    float* __restrict__ out)          // (32,1,31,63,63)
{
    __shared__ float LW[LW_SZ];
    __shared__ float LT[4][16][16];   // per-wave conv tile [M(spatial)][N(oc)]

    const int tid = threadIdx.x;

    // ---- repack weights into LDS: LW[kdkh][k][oc], k rows 9..11 zeroed ----
    for (int i = tid; i < LW_SZ; i += 128) {
        int oc   = i & 15;
        int k    = (i >> 4) % 12;
        int kdkh = i / (12 * 16);
        float v = 0.0f;
        if (k < 9) {
            int ic = k / 3, kw = k - 3 * ic;
            int kd = kdkh / 3, kh = kdkh - 3 * kd;
            v = w[ic * 432 + oc * 27 + kd * 9 + kh * 3 + kw];
        }
        LW[i] = v;
    }
    __syncthreads();

    const int oh = blockIdx.x;   // 0..62
    const int od = blockIdx.y;   // 0..30
    const int b  = blockIdx.z;   // 0..31

    const int wave = tid >> 5;
    const int lane = tid & 31;
    const int mrow = lane & 15;  // A: spatial row M; B/D: column N
    const int hi   = lane >> 4;  // 0 -> K slots {0,1}, 1 -> K slots {2,3}
    const int ow   = wave * 16 + mrow;

    // ---- per-thread A-gather table: independent of (kd,kh) ----
    // K slot for step s, half t:  k = 4s + 2*hi + t;  ic = k/3, kw = k%3
    int   aoff[6];
    float asel[6];
    #pragma unroll
    for (int j = 0; j < 6; ++j) {
        int s  = j >> 1, t = j & 1;
        int k  = s * 4 + hi * 2 + t;
        int ic = (k * 11) >> 5;          // k/3 (k <= 11)
        int kw = k - 3 * ic;
        int iw2 = ow + 1 - kw;
        int iw  = iw2 >> 1;
        bool ok = (k < 9) && ((iw2 & 1) == 0) && ((unsigned)iw < W_IN);
        aoff[j] = ok ? (ic * (D_IN * H_IN * W_IN) + iw) : 0;   // always in-bounds
        asel[j] = ok ? 1.0f : 0.0f;
    }

    // per-thread B base in LDS: element (kdkh*12 + k)*16 + mrow, k = 4s+2hi+t
    const float* lwlane = &LW[hi * 32 + mrow];

    v8f acc = {0.f, 0.f, 0.f, 0.f, 0.f, 0.f, 0.f, 0.f};

    const float* xb = x + (size_t)b * (IN_C * D_IN * H_IN * W_IN);

    for (int kd = 0; kd < 3; ++kd) {              // uniform: EXEC stays full
        int id2 = od + 1 - kd;
        if (id2 & 1) continue;
        int id = id2 >> 1;
        if ((unsigned)id >= D_IN) continue;
        for (int kh = 0; kh < 3; ++kh) {
            int ih2 = oh + 1 - kh;
            if (ih2 & 1) continue;
            int ih = ih2 >> 1;
            if ((unsigned)ih >= H_IN) continue;

            const float* xrow = xb + id * (H_IN * W_IN) + ih * W_IN; // uniform base
            const float* lw0  = lwlane + (kd * 3 + kh) * 192;

            // batch all 6 gathers (unconditional, masked by multiply)
            float a[6];
            #pragma unroll
            for (int j = 0; j < 6; ++j)
                a[j] = xrow[aoff[j]] * asel[j];

            #pragma unroll
            for (int s = 0; s < 3; ++s) {         // K = 12 -> 3 wmma steps
                v2f av, bv;
                av.x = a[2 * s];
                av.y = a[2 * s + 1];
                bv.x = lw0[s * 64];               // B[k][n]
                bv.y = lw0[s * 64 + 16];          // B[k+1][n]
                acc = __builtin_amdgcn_wmma_f32_16x16x4_f32(
                    false, av, false, bv, (short)0, acc, false, false);
            }
        }
    }

    // ---- spill D tile to LDS: lane holds N=mrow, VGPR v holds M=v+8*hi ----
    #pragma unroll
    for (int v = 0; v < 8; ++v) {
        LT[wave][v + hi * 8][mrow] = acc[v];
    }
    __syncthreads();

    // ---- channel logsumexp + hardswish + clip, 16 lanes per wave ----
    if (lane < 16) {
        const int M   = lane;
        const int owo = wave * 16 + M;
        if (owo < W_OUT) {
            float mx = LT[wave][M][0];
            #pragma unroll
            for (int c = 1; c < 16; ++c) mx = fmaxf(mx, LT[wave][M][c]);
            float ssum = 0.0f;
            #pragma unroll
            for (int c = 0; c < 16; ++c) ssum += __expf(LT[wave][M][c] - mx);
            float lse = mx + __logf(ssum);
            float sig = 1.0f / (1.0f + __expf(-(lse + 3.0f)));
            float hs  = lse * sig * (1.0f / 6.0f);
            float o   = hs - bias[0];
            o = fminf(1.0f, fmaxf(-1.0f, o));
            __builtin_nontemporal_store(
                o, &out[(((size_t)b * D_OUT + od) * H_OUT + oh) * W_OUT + owo]);
        }
    }
}

extern "C" void kernel_launch(void* const* d_in, const int* in_sizes, int n_in,
                              void* d_out, int out_size, void* d_ws, size_t ws_size,
                              hipStream_t stream) {
    const float* x    = (const float*)d_in[0];
    const float* w    = (const float*)d_in[1];
    const float* bias = (const float*)d_in[2];
    float* out        = (float*)d_out;
    (void)in_sizes; (void)n_in; (void)out_size; (void)d_ws; (void)ws_size;

    dim3 grid(H_OUT, D_OUT, 32);   // (oh, od, b)
    dim3 block(128);               // 4 waves x 16 ow each
    convt_lse_hswish_kernel<<<grid, block, 0, stream>>>(x, w, bias, out);
}